// AttentionEdges_29618094473623
// MI455X (gfx1250) — compile-verified
//
#include <hip/hip_runtime.h>
#include <math.h>
#include <stdint.h>

#define N_EDGE 100000
#define T_PAIR 300000
#define HSCA 128
#define HVEC 32
#define KSCA 64
#define KVEC 16
#define NHEAD 4
#define NGAUSS 125
#define CATDIM (HVEC + HSCA)   // 160

__device__ __forceinline__ float gcoeff() {
    // -0.5 / (CUTOFF/(NG-1))^2 with CUTOFF=10, NG=125
    return -0.5f / ((10.0f / 124.0f) * (10.0f / 124.0f));
}

typedef float v2f __attribute__((ext_vector_type(2)));
typedef float v4f __attribute__((ext_vector_type(4)));
typedef float v8f __attribute__((ext_vector_type(8)));

// ---- explicit global (AS1) accessors: guarantee global_load/store, not flat ----
typedef __attribute__((address_space(1))) float*          gfptr;
typedef const __attribute__((address_space(1))) float*    gcfptr;
typedef const __attribute__((address_space(1))) v2f*      gc2ptr;
typedef const __attribute__((address_space(1))) v4f*      gc4ptr;
typedef __attribute__((address_space(1))) unsigned*       guptr;
typedef const __attribute__((address_space(1))) unsigned* gcuptr;
typedef const __attribute__((address_space(1))) int*      gciptr;

__device__ __forceinline__ float ldg(const float* p)  { return *(gcfptr)(uintptr_t)p; }
__device__ __forceinline__ v2f   ldg2(const float* p) { return *(gc2ptr)(uintptr_t)p; }
__device__ __forceinline__ v4f   ldg4(const float* p) { return *(gc4ptr)(uintptr_t)p; }
__device__ __forceinline__ unsigned ldgu(const unsigned* p) { return *(gcuptr)(uintptr_t)p; }
__device__ __forceinline__ int   ldgi(const int* p)   { return *(gciptr)(uintptr_t)p; }
__device__ __forceinline__ void  stg(float* p, float v) { *(gfptr)(uintptr_t)p = v; }

__device__ __forceinline__ void gatomic_add(float* p, float v) {
    __hip_atomic_fetch_add((gfptr)(uintptr_t)p, v, __ATOMIC_RELAXED, __HIP_MEMORY_SCOPE_AGENT);
}
__device__ __forceinline__ void gatomic_maxu(unsigned* p, unsigned v) {
    __hip_atomic_fetch_max((guptr)(uintptr_t)p, v, __ATOMIC_RELAXED, __HIP_MEMORY_SCOPE_AGENT);
}

__device__ __forceinline__ v8f v8zero() {
    v8f z = {0.f, 0.f, 0.f, 0.f, 0.f, 0.f, 0.f, 0.f};
    return z;
}

__device__ __forceinline__ v8f wmma_f32(v2f a, v2f b, v8f c) {
    // V_WMMA_F32_16X16X4_F32 : D = A(16x4,f32) x B(4x16,f32) + C(16x16,f32)
    return __builtin_amdgcn_wmma_f32_16x16x4_f32(false, a, false, b, (short)0, c, false, false);
}

// ---------------------------------------------------------------------------
// GVLinear projection for a 16-edge tile, one wave (wave32).
// A-frag element (vgpr p): m = lane%16, k = kb + p + 2*(lane/16)
// B-frag element (vgpr p): n = lane%16, k = kb + p + 2*(lane/16)
// C-frag element (vgpr r): n = lane%16 (+16*nt), m = r + 8*(lane/16)
// LDS per wave: interT[3][32][16] | nrmT[32][16] | oscaT[OS][16]  (<=4096 f)
// Must be inlined so the __shared__ provenance survives -> DS ops, and global
// pointers stay recognizable -> global_* ops.
// ---------------------------------------------------------------------------
template <int OS, int OV>
__device__ __attribute__((always_inline)) void gv_project(
        int n0,
        const float* __restrict__ esca,
        const float* __restrict__ evec,
        const float* __restrict__ Wv1,
        const float* __restrict__ Wv2,
        const float* __restrict__ Ws,
        const float* __restrict__ Wg,
        const float* __restrict__ bg,
        float* __restrict__ osca_g,
        float* __restrict__ ovec_g,
        float* __restrict__ lds) {
    const int lane = threadIdx.x & 31;
    const int lo = lane & 15;
    const int hi = lane >> 4;
    float* interT = lds;                         // 1536
    float* nrmT   = lds + 3 * 32 * 16;           // 512
    float* oscaT  = lds + 3 * 32 * 16 + 32 * 16; // up to 2048

    int rowA = n0 + lo;
    if (rowA >= N_EDGE) rowA = N_EDGE - 1;

    // ---- Stage 1: inter[o,d] = sum_c Wv1[o,c] * evec[m,c,d] ----
    v8f sumsq[2];
    sumsq[0] = v8zero();
    sumsq[1] = v8zero();
    for (int d = 0; d < 3; ++d) {
        for (int nt = 0; nt < 2; ++nt) {
            v8f acc = v8zero();
#pragma unroll
            for (int kt = 0; kt < 8; ++kt) {
                const int kb = kt * 4 + 2 * hi;
                v2f a;
                a.x = ldg(&evec[(size_t)rowA * (HVEC * 3) + (kb + 0) * 3 + d]);
                a.y = ldg(&evec[(size_t)rowA * (HVEC * 3) + (kb + 1) * 3 + d]);
                const v2f b = ldg2(&Wv1[(nt * 16 + lo) * HVEC + kb]);
                acc = wmma_f32(a, b, acc);
            }
            sumsq[nt] += acc * acc;
#pragma unroll
            for (int r = 0; r < 8; ++r)
                interT[(d * 32 + nt * 16 + lo) * 16 + (r + 8 * hi)] = acc[r];
        }
    }
#pragma unroll
    for (int nt = 0; nt < 2; ++nt)
#pragma unroll
        for (int r = 0; r < 8; ++r)
            nrmT[(nt * 16 + lo) * 16 + (r + 8 * hi)] = sqrtf(sumsq[nt][r]);

    // ---- Stage 2: out_sca = [nrm | sca] @ Ws^T  (K = 160) ----
    v8f osca[OS / 16];
    for (int nt = 0; nt < OS / 16; ++nt) {
        v8f acc = v8zero();
#pragma unroll
        for (int kt = 0; kt < 8; ++kt) {                    // k in [0,32): nrm part (LDS)
            const int kb = kt * 4 + 2 * hi;
            v2f a;
            a.x = nrmT[(kb + 0) * 16 + lo];
            a.y = nrmT[(kb + 1) * 16 + lo];
            const v2f b = ldg2(&Ws[(nt * 16 + lo) * CATDIM + kb]);
            acc = wmma_f32(a, b, acc);
        }
        for (int kt = 8; kt < 40; ++kt) {                   // k in [32,160): sca part (b64 loads)
            const int kb = kt * 4 + 2 * hi;
            const v2f a = ldg2(&esca[(size_t)rowA * HSCA + (kb - HVEC)]);
            const v2f b = ldg2(&Ws[(nt * 16 + lo) * CATDIM + kb]);
            acc = wmma_f32(a, b, acc);
        }
        osca[nt] = acc;
#pragma unroll
        for (int r = 0; r < 8; ++r) {
            const int m = r + 8 * hi, o = nt * 16 + lo;
            oscaT[o * 16 + m] = acc[r];
            if (n0 + m < N_EDGE) stg(&osca_g[(size_t)(n0 + m) * OS + o], acc[r]);
        }
    }

    // ---- Stage 3: out_vec[o,d] = sum_c Wv2[o,c] * inter[c,d] ----
    v8f ovec[3][OV / 16];
    for (int d = 0; d < 3; ++d)
        for (int nt = 0; nt < OV / 16; ++nt) {
            v8f acc = v8zero();
#pragma unroll
            for (int kt = 0; kt < 8; ++kt) {
                const int kb = kt * 4 + 2 * hi;
                v2f a;
                a.x = interT[(d * 32 + kb + 0) * 16 + lo];
                a.y = interT[(d * 32 + kb + 1) * 16 + lo];
                const v2f b = ldg2(&Wv2[(nt * 16 + lo) * HVEC + kb]);
                acc = wmma_f32(a, b, acc);
            }
            ovec[d][nt] = acc;
        }

    // ---- Stage 4: gate = sigmoid(out_sca @ Wg^T + bg) ----
    v8f gate[OV / 16];
    for (int nt = 0; nt < OV / 16; ++nt) {
        v8f acc = v8zero();
        for (int kt = 0; kt < OS / 4; ++kt) {
            const int kb = kt * 4 + 2 * hi;
            v2f a;
            a.x = oscaT[(kb + 0) * 16 + lo];
            a.y = oscaT[(kb + 1) * 16 + lo];
            const v2f b = ldg2(&Wg[(nt * 16 + lo) * OS + kb]);
            acc = wmma_f32(a, b, acc);
        }
        const float bgv = ldg(&bg[nt * 16 + lo]);
#pragma unroll
        for (int r = 0; r < 8; ++r)
            gate[nt][r] = 1.0f / (1.0f + __expf(-(acc[r] + bgv)));
    }

    // ---- Stage 5: store gated vectors ----
    for (int d = 0; d < 3; ++d)
        for (int nt = 0; nt < OV / 16; ++nt)
#pragma unroll
            for (int r = 0; r < 8; ++r) {
                const int m = r + 8 * hi, o = nt * 16 + lo;
                if (n0 + m < N_EDGE)
                    stg(&ovec_g[((size_t)(n0 + m) * OV + o) * 3 + d],
                        gate[nt][r] * ovec[d][nt][r]);
            }
}

// ---------------------------------------------------------------------------
__global__ __launch_bounds__(128) void qkv_kernel(
    const float* __restrict__ esca, const float* __restrict__ evec,
    const float* qWv1, const float* qWv2, const float* qWs, const float* qWg, const float* qbg,
    const float* kWv1, const float* kWv2, const float* kWs, const float* kWg, const float* kbg,
    const float* vWv1, const float* vWv2, const float* vWs, const float* vWg, const float* vbg,
    float* q_sca, float* q_vec, float* k_sca, float* k_vec, float* v_sca, float* v_vec) {
    __shared__ float smem[4 * 4096];
    const int wave = threadIdx.x >> 5;
    const int tile = blockIdx.x * 4 + wave;
    if (tile * 16 >= N_EDGE) return;
    float* lds = smem + wave * 4096;
    const int n0 = tile * 16;
    // prefetch this tile's features (global_prefetch_b8)
    __builtin_prefetch(esca + (size_t)n0 * HSCA, 0, 3);
    __builtin_prefetch(evec + (size_t)n0 * (HVEC * 3), 0, 3);
    gv_project<KSCA, KVEC>(n0, esca, evec, qWv1, qWv2, qWs, qWg, qbg, q_sca, q_vec, lds);
    gv_project<KSCA, KVEC>(n0, esca, evec, kWv1, kWv2, kWs, kWg, kbg, k_sca, k_vec, lds);
    gv_project<HSCA, HVEC>(n0, esca, evec, vWv1, vWv2, vWs, vWg, vbg, v_sca, v_vec, lds);
}

// ---------------------------------------------------------------------------
// Bias GVLinear over pairs. vec_feat is rank-1 => inter[o,d]=w1v[o]*unit[d].
// b_Ws (4x160) and gauss offsets staged in LDS, reused by all 256 threads.
// ---------------------------------------------------------------------------
__global__ __launch_bounds__(256) void bias_kernel(
    const float* __restrict__ pos, const float* __restrict__ trif,
    const int* __restrict__ tri_idx,
    const float* __restrict__ bWv1, const float* __restrict__ bWv2,
    const float* __restrict__ bWs, const float* __restrict__ bWg,
    const float* __restrict__ bbg, const float* __restrict__ vexp,
    const float* __restrict__ goff,
    float* __restrict__ bias_sca, float* __restrict__ bias_vec) {
    __shared__ float w1v[HVEC];
    __shared__ float w2v[NHEAD];
    __shared__ float sWs[NHEAD * CATDIM];
    __shared__ float sgoff[NGAUSS];
    const int tid = threadIdx.x;
    for (int c = tid; c < NHEAD * CATDIM; c += 256) sWs[c] = ldg(&bWs[c]);
    for (int g = tid; g < NGAUSS; g += 256) sgoff[g] = ldg(&goff[g]);
    if (tid < HVEC) {
        float s = 0.f;
        for (int c = 0; c < HVEC; ++c) s += ldg(&bWv1[tid * HVEC + c]) * ldg(&vexp[c]);
        w1v[tid] = s;
    }
    __syncthreads();
    if (tid < NHEAD) {
        float s = 0.f;
        for (int c = 0; c < HVEC; ++c) s += ldg(&bWv2[tid * HVEC + c]) * w1v[c];
        w2v[tid] = s;
    }
    __syncthreads();

    const int t = blockIdx.x * 256 + tid;
    if (t >= T_PAIR) return;
    const int na = ldgi(&tri_idx[t]), nb = ldgi(&tri_idx[T_PAIR + t]);
    const float vx = ldg(&pos[na * 3 + 0]) - ldg(&pos[nb * 3 + 0]);
    const float vy = ldg(&pos[na * 3 + 1]) - ldg(&pos[nb * 3 + 1]);
    const float vz = ldg(&pos[na * 3 + 2]) - ldg(&pos[nb * 3 + 2]);
    const float dist = sqrtf(vx * vx + vy * vy + vz * vz);
    const float inv = 1.0f / (dist + 1e-7f);
    const float unorm = dist * inv;          // |unit|
    const float unorm2 = unorm * unorm;

    float os[NHEAD] = {0.f, 0.f, 0.f, 0.f};
    for (int c = 0; c < HVEC; ++c) {
        const float nr = fabsf(w1v[c]) * unorm;
#pragma unroll
        for (int h = 0; h < NHEAD; ++h) os[h] += sWs[h * CATDIM + c] * nr;
    }
    const float gc = gcoeff();
    for (int g = 0; g < NGAUSS; ++g) {
        const float dd = dist - sgoff[g];
        const float f = __expf(gc * dd * dd);
#pragma unroll
        for (int h = 0; h < NHEAD; ++h) os[h] += sWs[h * CATDIM + HVEC + g] * f;
    }
#pragma unroll
    for (int e = 0; e < 3; ++e) {
        const float f = ldg(&trif[t * 3 + e]);
#pragma unroll
        for (int h = 0; h < NHEAD; ++h) os[h] += sWs[h * CATDIM + HVEC + NGAUSS + e] * f;
    }
#pragma unroll
    for (int h = 0; h < NHEAD; ++h) {
        float x = ldg(&bbg[h]);
#pragma unroll
        for (int h2 = 0; h2 < NHEAD; ++h2) x += ldg(&bWg[h * NHEAD + h2]) * os[h2];
        const float gate = 1.0f / (1.0f + __expf(-x));
        const float gw = gate * w2v[h];
        stg(&bias_sca[(size_t)t * NHEAD + h], os[h]);
        stg(&bias_vec[(size_t)t * NHEAD + h], gw * gw * unorm2); // sum_d (gate*w2v*unit_d)^2
    }
}

// ---------------------------------------------------------------------------
// float <-> order-preserving uint key (for atomic max)
// ---------------------------------------------------------------------------
__device__ __forceinline__ unsigned fkey(float x) {
    const unsigned u = __float_as_uint(x);
    return (u & 0x80000000u) ? ~u : (u | 0x80000000u);
}
__device__ __forceinline__ float fdecode(unsigned k) {
    return (k & 0x80000000u) ? __uint_as_float(k & 0x7FFFFFFFu) : __uint_as_float(~k);
}

__global__ __launch_bounds__(256) void zero_kernel(float* p, int n) {
    const int i = blockIdx.x * 256 + threadIdx.x;
    if (i < n) stg(&p[i], 0.f);
}

// ---- pass 1: logits + running max --------------------------------------
__global__ __launch_bounds__(256) void logits_kernel(
    const float* __restrict__ q_sca, const float* __restrict__ q_vec,
    const float* __restrict__ k_sca, const float* __restrict__ k_vec,
    const int* __restrict__ idx_i, const int* __restrict__ idx_j,
    const float* __restrict__ bias_sca, const float* __restrict__ bias_vec,
    float* __restrict__ l_sca, float* __restrict__ l_vec,
    unsigned* __restrict__ max_sca, unsigned* __restrict__ max_vec) {
    const int t = blockIdx.x * 256 + threadIdx.x;
    if (t >= T_PAIR) return;
    const int i = ldgi(&idx_i[t]), j = ldgi(&idx_j[t]);
#pragma unroll
    for (int h = 0; h < NHEAD; ++h) {
        float s = 0.f;
#pragma unroll
        for (int c4 = 0; c4 < 4; ++c4) {   // 16 floats = 4 x b128
            const v4f qa = ldg4(&q_sca[(size_t)i * KSCA + h * 16 + c4 * 4]);
            const v4f ka = ldg4(&k_sca[(size_t)j * KSCA + h * 16 + c4 * 4]);
#pragma unroll
            for (int e = 0; e < 4; ++e) s += qa[e] * ka[e];
        }
        float v = 0.f;
#pragma unroll
        for (int c4 = 0; c4 < 3; ++c4) {   // 12 floats = 3 x b128
            const v4f qa = ldg4(&q_vec[(size_t)i * (KVEC * 3) + h * 12 + c4 * 4]);
            const v4f ka = ldg4(&k_vec[(size_t)j * (KVEC * 3) + h * 12 + c4 * 4]);
#pragma unroll
            for (int e = 0; e < 4; ++e) v += qa[e] * ka[e];
        }
        const float ls = ldg(&bias_sca[(size_t)t * NHEAD + h]) + s;
        const float lv = ldg(&bias_vec[(size_t)t * NHEAD + h]) + v;
        stg(&l_sca[(size_t)t * NHEAD + h], ls);
        stg(&l_vec[(size_t)t * NHEAD + h], lv);
        gatomic_maxu(&max_sca[i * NHEAD + h], fkey(ls));
        gatomic_maxu(&max_vec[i * NHEAD + h], fkey(lv));
    }
}

// ---- pass 2: exp + running sum ----------------------------------------
__global__ __launch_bounds__(256) void expsum_kernel(
    const int* __restrict__ idx_i,
    float* __restrict__ l_sca, float* __restrict__ l_vec,
    const unsigned* __restrict__ max_sca, const unsigned* __restrict__ max_vec,
    float* __restrict__ sum_sca, float* __restrict__ sum_vec) {
    const int gid = blockIdx.x * 256 + threadIdx.x;
    if (gid >= T_PAIR * NHEAD) return;
    const int t = gid >> 2, h = gid & 3;
    const int i = ldgi(&idx_i[t]);
    const float es = __expf(ldg(&l_sca[gid]) - fdecode(ldgu(&max_sca[i * NHEAD + h])));
    const float ev = __expf(ldg(&l_vec[gid]) - fdecode(ldgu(&max_vec[i * NHEAD + h])));
    stg(&l_sca[gid], es);
    stg(&l_vec[gid], ev);
    gatomic_add(&sum_sca[i * NHEAD + h], es);
    gatomic_add(&sum_vec[i * NHEAD + h], ev);
}

// ---- pass 3: alpha * v scatter-add ------------------------------------
__global__ __launch_bounds__(256) void agg_kernel(
    const int* __restrict__ idx_i, const int* __restrict__ idx_j,
    const float* __restrict__ l_sca, const float* __restrict__ l_vec,
    const float* __restrict__ sum_sca, const float* __restrict__ sum_vec,
    const float* __restrict__ v_sca, const float* __restrict__ v_vec,
    float* __restrict__ acc_sca, float* __restrict__ acc_vec) {
    const int gid = blockIdx.x * 256 + threadIdx.x;
    if (gid >= T_PAIR * NHEAD) return;
    const int t = gid >> 2, h = gid & 3;
    const int i = ldgi(&idx_i[t]), j = ldgi(&idx_j[t]);
    const float as = ldg(&l_sca[gid]) / ldg(&sum_sca[i * NHEAD + h]);
    const float av = ldg(&l_vec[gid]) / ldg(&sum_vec[i * NHEAD + h]);
#pragma unroll
    for (int c4 = 0; c4 < 8; ++c4) {       // 32 floats = 8 x b128 gathers
        const v4f vv = ldg4(&v_sca[(size_t)j * HSCA + h * 32 + c4 * 4]);
#pragma unroll
        for (int e = 0; e < 4; ++e)
            gatomic_add(&acc_sca[(size_t)i * HSCA + h * 32 + c4 * 4 + e], as * vv[e]);
    }
#pragma unroll
    for (int c4 = 0; c4 < 6; ++c4) {       // 24 floats = 6 x b128 gathers
        const v4f vv = ldg4(&v_vec[(size_t)j * (HVEC * 3) + h * 24 + c4 * 4]);
#pragma unroll
        for (int e = 0; e < 4; ++e)
            gatomic_add(&acc_vec[(size_t)i * (HVEC * 3) + h * 24 + c4 * 4 + e], av * vv[e]);
    }
}

// ---- finalize: residual + LayerNorm, one wave per edge ----------------
__global__ __launch_bounds__(256) void ln_kernel(
    const float* __restrict__ esca, const float* __restrict__ evec,
    const float* __restrict__ acc_sca, const float* __restrict__ acc_vec,
    const float* __restrict__ lsw, const float* __restrict__ lsb,
    const float* __restrict__ lvw, const float* __restrict__ lvb,
    float* __restrict__ out) {
    const int wave = threadIdx.x >> 5, lane = threadIdx.x & 31;
    const int n = blockIdx.x * 8 + wave;
    if (n >= N_EDGE) return;
    // scalar LN over 128 (4 per lane, b128 loads)
    const v4f e4 = ldg4(&esca[(size_t)n * HSCA + lane * 4]);
    const v4f a4 = ldg4(&acc_sca[(size_t)n * HSCA + lane * 4]);
    float x[4], s = 0.f;
#pragma unroll
    for (int q = 0; q < 4; ++q) { x[q] = e4[q] + a4[q]; s += x[q]; }
#pragma unroll
    for (int off = 16; off >= 1; off >>= 1) s += __shfl_xor(s, off, 32);
    const float mu = s * (1.0f / 128.0f);
    float var = 0.f;
#pragma unroll
    for (int q = 0; q < 4; ++q) { const float d = x[q] - mu; var += d * d; }
#pragma unroll
    for (int off = 16; off >= 1; off >>= 1) var += __shfl_xor(var, off, 32);
    const float inv = rsqrtf(var * (1.0f / 128.0f) + 1e-5f);
#pragma unroll
    for (int q = 0; q < 4; ++q) {
        const int c = lane * 4 + q;
        stg(&out[(size_t)n * HSCA + c], (x[q] - mu) * inv * ldg(&lsw[c]) + ldg(&lsb[c]));
    }
    // vector LN over 96 (3 per lane, ch = lane)
    float y[3], s2 = 0.f;
#pragma unroll
    for (int d = 0; d < 3; ++d) {
        y[d] = ldg(&evec[(size_t)n * 96 + lane * 3 + d]) + ldg(&acc_vec[(size_t)n * 96 + lane * 3 + d]);
        s2 += y[d];
    }
#pragma unroll
    for (int off = 16; off >= 1; off >>= 1) s2 += __shfl_xor(s2, off, 32);
    const float mu2 = s2 * (1.0f / 96.0f);
    float var2 = 0.f;
#pragma unroll
    for (int d = 0; d < 3; ++d) { const float dd = y[d] - mu2; var2 += dd * dd; }
#pragma unroll
    for (int off = 16; off >= 1; off >>= 1) var2 += __shfl_xor(var2, off, 32);
    const float inv2 = rsqrtf(var2 * (1.0f / 96.0f) + 1e-5f);
#pragma unroll
    for (int d = 0; d < 3; ++d)
        stg(&out[(size_t)N_EDGE * HSCA + (size_t)n * 96 + lane * 3 + d],
            (y[d] - mu2) * inv2 * ldg(&lvw[lane * 3 + d]) + ldg(&lvb[lane * 3 + d]));
}

// ---------------------------------------------------------------------------
extern "C" void kernel_launch(void* const* d_in, const int* in_sizes, int n_in,
                              void* d_out, int out_size, void* d_ws, size_t ws_size,
                              hipStream_t stream) {
    (void)in_sizes; (void)n_in; (void)out_size; (void)ws_size;
    const float* edge_sca = (const float*)d_in[0];
    const float* edge_vec = (const float*)d_in[1];
    const float* pos      = (const float*)d_in[2];
    const float* trif     = (const float*)d_in[3];
    const float* qWv1 = (const float*)d_in[4],  *qWv2 = (const float*)d_in[5];
    const float* qWs  = (const float*)d_in[6],  *qWg  = (const float*)d_in[7],  *qbg = (const float*)d_in[8];
    const float* kWv1 = (const float*)d_in[9],  *kWv2 = (const float*)d_in[10];
    const float* kWs  = (const float*)d_in[11], *kWg  = (const float*)d_in[12], *kbg = (const float*)d_in[13];
    const float* vWv1 = (const float*)d_in[14], *vWv2 = (const float*)d_in[15];
    const float* vWs  = (const float*)d_in[16], *vWg  = (const float*)d_in[17], *vbg = (const float*)d_in[18];
    const float* bWv1 = (const float*)d_in[19], *bWv2 = (const float*)d_in[20];
    const float* bWs  = (const float*)d_in[21], *bWg  = (const float*)d_in[22], *bbg = (const float*)d_in[23];
    const float* vexp = (const float*)d_in[24], *goff = (const float*)d_in[25];
    const float* lsw = (const float*)d_in[26], *lsb = (const float*)d_in[27];
    const float* lvw = (const float*)d_in[28], *lvb = (const float*)d_in[29];
    const int* idx_i = (const int*)d_in[31];
    const int* idx_j = (const int*)d_in[32];
    const int* tri_idx = (const int*)d_in[33];
    float* out = (float*)d_out;

    // workspace carve (floats)
    float* ws = (float*)d_ws;
    size_t o = 0;
    float* q_sca = ws + o; o += (size_t)N_EDGE * KSCA;
    float* q_vec = ws + o; o += (size_t)N_EDGE * KVEC * 3;
    float* k_sca = ws + o; o += (size_t)N_EDGE * KSCA;
    float* k_vec = ws + o; o += (size_t)N_EDGE * KVEC * 3;
    float* v_sca = ws + o; o += (size_t)N_EDGE * HSCA;
    float* v_vec = ws + o; o += (size_t)N_EDGE * HVEC * 3;
    float* bias_sca = ws + o; o += (size_t)T_PAIR * NHEAD;
    float* bias_vec = ws + o; o += (size_t)T_PAIR * NHEAD;
    float* l_sca = ws + o; o += (size_t)T_PAIR * NHEAD;
    float* l_vec = ws + o; o += (size_t)T_PAIR * NHEAD;
    float* zbase = ws + o;                                   // zeroed region start
    unsigned* max_sca = (unsigned*)(ws + o); o += (size_t)N_EDGE * NHEAD;
    unsigned* max_vec = (unsigned*)(ws + o); o += (size_t)N_EDGE * NHEAD;
    float* sum_sca = ws + o; o += (size_t)N_EDGE * NHEAD;
    float* sum_vec = ws + o; o += (size_t)N_EDGE * NHEAD;
    float* acc_sca = ws + o; o += (size_t)N_EDGE * HSCA;
    float* acc_vec = ws + o; o += (size_t)N_EDGE * HVEC * 3;

    const int zn = N_EDGE * NHEAD * 4 + N_EDGE * (HSCA + HVEC * 3); // 24e6
    zero_kernel<<<(zn + 255) / 256, 256, 0, stream>>>(zbase, zn);

    const int tiles = (N_EDGE + 15) / 16;
    qkv_kernel<<<(tiles + 3) / 4, 128, 0, stream>>>(
        edge_sca, edge_vec,
        qWv1, qWv2, qWs, qWg, qbg,
        kWv1, kWv2, kWs, kWg, kbg,
        vWv1, vWv2, vWs, vWg, vbg,
        q_sca, q_vec, k_sca, k_vec, v_sca, v_vec);

    bias_kernel<<<(T_PAIR + 255) / 256, 256, 0, stream>>>(
        pos, trif, tri_idx, bWv1, bWv2, bWs, bWg, bbg, vexp, goff,
        bias_sca, bias_vec);

    logits_kernel<<<(T_PAIR + 255) / 256, 256, 0, stream>>>(
        q_sca, q_vec, k_sca, k_vec, idx_i, idx_j,
        bias_sca, bias_vec, l_sca, l_vec, max_sca, max_vec);

    expsum_kernel<<<(T_PAIR * NHEAD + 255) / 256, 256, 0, stream>>>(
        idx_i, l_sca, l_vec, max_sca, max_vec, sum_sca, sum_vec);

    agg_kernel<<<(T_PAIR * NHEAD + 255) / 256, 256, 0, stream>>>(
        idx_i, idx_j, l_sca, l_vec, sum_sca, sum_vec, v_sca, v_vec,
        acc_sca, acc_vec);

    ln_kernel<<<(N_EDGE + 7) / 8, 256, 0, stream>>>(
        edge_sca, edge_vec, acc_sca, acc_vec, lsw, lsb, lvw, lvb, out);
}